// Model_test_add_50869592655498
// MI455X (gfx1250) — compile-verified
//
#include <hip/hip_runtime.h>

// ---------------------------------------------------------------------------
// Sizes fixed by the reference model
#define NN_H      128
#define NN_D      256
#define NN_B      64
#define NN_T      30     // NUM_TYPES
#define NN_TP     32     // padded types (K multiple of 4)
#define NN_OUT    10
#define BN_EPS    1e-5f

typedef float v2f __attribute__((ext_vector_type(2)));
typedef float v8f __attribute__((ext_vector_type(8)));

// ---------------------------------------------------------------------------
__global__ __launch_bounds__(256) void k_zero(int* p, int n) {
    int i = blockIdx.x * blockDim.x + threadIdx.x;
    if (i < n) p[i] = 0;
}

// cnt[v] = multiplicity of v in cycle_vertex
__global__ __launch_bounds__(256) void k_count(const int* __restrict__ cv,
                                               int* __restrict__ cnt, int E) {
    int i = blockIdx.x * blockDim.x + threadIdx.x;
    if (i < E) atomicAdd(&cnt[cv[i]], 1);
}

// C[b,t] += cnt[v] for batch[v]=b, x[v]=t
__global__ __launch_bounds__(256) void k_hist(const int* __restrict__ x,
                                              const int* __restrict__ batch,
                                              const int* __restrict__ cnt,
                                              int* __restrict__ C, int N) {
    int i = blockIdx.x * blockDim.x + threadIdx.x;
    if (i < N) {
        int c = cnt[i];
        if (c) atomicAdd(&C[batch[i] * NN_T + x[i]], c);
    }
}

// ---------------------------------------------------------------------------
// Whole dense chain in one workgroup (8 waves), fp32 WMMA 16x16x4.
// Fragment layouts per CDNA5 ISA 7.12.2:
//  A 16x4 : lane<16 -> M=lane, K={k0,k0+1}; lane>=16 -> M=lane-16, K={k0+2,k0+3}
//  B 4x16 : lane<16 -> N=lane, K rows {k0,k0+1}; lane>=16 -> K rows {k0+2,k0+3}
//  C/D    : N=lane%16; lane<16 rows m0..m0+7 ; lane>=16 rows m0+8..m0+15
__global__ __launch_bounds__(256) void k_dense(
    const int*   __restrict__ Cint,   // [64,30]
    const float* __restrict__ emb,    // [30,128]
    const float* __restrict__ Wp,     // [128,128]
    const float* __restrict__ bp,     // [128]
    const float* __restrict__ W1,     // [128,256]
    const float* __restrict__ g1,     // [256]
    const float* __restrict__ be1,    // [256]
    const float* __restrict__ W2,     // [256,256]
    const float* __restrict__ g2,     // [256]
    const float* __restrict__ be2,    // [256]
    const float* __restrict__ W3,     // [256,10]
    const float* __restrict__ b3,     // [10]
    float* __restrict__ g_ws,         // [64,128] scratch (global)
    float* __restrict__ z2_ws,        // [64,256] scratch (global)
    float* __restrict__ out)          // [64,10]
{
    __shared__ float smem[16384];                 // 64 KB, regions aliased by phase
    float* Cf   = smem;                           // [64,32]   8 KB   (phase A)
    float* embp = smem + 2048;                    // [32,128] 16 KB   (phase A)
    float* S    = smem + 6144;                    // [64,128] 32 KB   (phase A)
    float* Tl   = smem + 14336;                   // [64]             (phase A)
    float* z1   = smem;                           // [64,256] 64 KB   (phase B)

    const int tid  = threadIdx.x;
    const int lane = tid & 31;
    const int wave = tid >> 5;
    const int half = lane >> 4;       // 0|1
    const int lm   = lane & 15;
    const int koff = half * 2;        // A/B fragment K offset for upper half-wave
    const int moff = half * 8;        // C/D row offset for upper half-wave

    // ---- stage 0: stage count matrix (float, K-padded) + padded embedding ----
    for (int i = tid; i < NN_B * NN_TP; i += 256) {
        int b = i >> 5, t = i & 31;
        Cf[i] = (t < NN_T) ? (float)Cint[b * NN_T + t] : 0.0f;
    }
    for (int i = tid; i < NN_TP * NN_H; i += 256)
        embp[i] = (i < NN_T * NN_H) ? emb[i] : 0.0f;
    if (tid < NN_B) {
        int s = 0;
        for (int t = 0; t < NN_T; ++t) s += Cint[tid * NN_T + t];
        Tl[tid] = (float)s;
    }
    __syncthreads();

    // ---- stage 1: S[64,128] = Cf[64,32] @ embp[32,128]  (LDS x LDS) ----
    for (int t = wave; t < 32; t += 8) {
        int m0 = (t >> 3) * 16, n0 = (t & 7) * 16;
        v8f acc = {0, 0, 0, 0, 0, 0, 0, 0};
        for (int k0 = 0; k0 < NN_TP; k0 += 4) {
            const float* ap = &Cf[(m0 + lm) * NN_TP + k0 + koff];
            const float* bq = &embp[(k0 + koff) * NN_H + n0 + lm];
            v2f a = {ap[0], ap[1]};
            v2f b = {bq[0], bq[NN_H]};
            acc = __builtin_amdgcn_wmma_f32_16x16x4_f32(false, a, false, b,
                                                        (short)0, acc, false, false);
        }
        float* dp = &S[(m0 + moff) * NN_H + n0 + lm];
#pragma unroll
        for (int r = 0; r < 8; ++r) dp[r * NN_H] = acc[r];
    }
    __syncthreads();

    // ---- stage 2: g[64,128] = S @ Wp[128,128] ; then g += T[m]*bp[n] ----
    for (int t = wave; t < 32; t += 8) {
        int m0 = (t >> 3) * 16, n0 = (t & 7) * 16;
        v8f acc = {0, 0, 0, 0, 0, 0, 0, 0};
        for (int k0 = 0; k0 < NN_H; k0 += 4) {
            const float* ap = &S[(m0 + lm) * NN_H + k0 + koff];
            const float* bq = &Wp[(k0 + koff) * NN_H + n0 + lm];
            v2f a = {ap[0], ap[1]};
            v2f b = {bq[0], bq[NN_H]};
            acc = __builtin_amdgcn_wmma_f32_16x16x4_f32(false, a, false, b,
                                                        (short)0, acc, false, false);
        }
        float* dp = &g_ws[(m0 + moff) * NN_H + n0 + lm];
#pragma unroll
        for (int r = 0; r < 8; ++r) dp[r * NN_H] = acc[r];
    }
    __syncthreads();
    for (int i = tid; i < NN_B * NN_H; i += 256)
        g_ws[i] += Tl[i >> 7] * bp[i & (NN_H - 1)];
    __syncthreads();

    // ---- stage 3: z1[64,256] = g @ W1[128,256] ; BN1+ReLU (b1 cancels in BN) ----
    for (int t = wave; t < 64; t += 8) {
        int m0 = (t >> 4) * 16, n0 = (t & 15) * 16;
        v8f acc = {0, 0, 0, 0, 0, 0, 0, 0};
        for (int k0 = 0; k0 < NN_H; k0 += 4) {
            const float* ap = &g_ws[(m0 + lm) * NN_H + k0 + koff];
            const float* bq = &W1[(k0 + koff) * NN_D + n0 + lm];
            v2f a = {ap[0], ap[1]};
            v2f b = {bq[0], bq[NN_D]};
            acc = __builtin_amdgcn_wmma_f32_16x16x4_f32(false, a, false, b,
                                                        (short)0, acc, false, false);
        }
        float* dp = &z1[(m0 + moff) * NN_D + n0 + lm];
#pragma unroll
        for (int r = 0; r < 8; ++r) dp[r * NN_D] = acc[r];
    }
    __syncthreads();
    {   // BN over 64 rows, biased variance, then ReLU (column j = tid)
        int j = tid;
        float s = 0.f, sq = 0.f;
        for (int i = 0; i < NN_B; ++i) { float v = z1[i * NN_D + j]; s += v; sq += v * v; }
        float mean = s * (1.0f / NN_B);
        float var  = sq * (1.0f / NN_B) - mean * mean;
        float sc   = g1[j] / sqrtf(var + BN_EPS);
        float be   = be1[j];
        for (int i = 0; i < NN_B; ++i) {
            float v = (z1[i * NN_D + j] - mean) * sc + be;
            z1[i * NN_D + j] = v > 0.f ? v : 0.f;
        }
    }
    __syncthreads();

    // ---- stage 4: z2[64,256] = z1 @ W2[256,256] ; BN2+ReLU ----
    for (int t = wave; t < 64; t += 8) {
        int m0 = (t >> 4) * 16, n0 = (t & 15) * 16;
        v8f acc = {0, 0, 0, 0, 0, 0, 0, 0};
        for (int k0 = 0; k0 < NN_D; k0 += 4) {
            const float* ap = &z1[(m0 + lm) * NN_D + k0 + koff];
            const float* bq = &W2[(k0 + koff) * NN_D + n0 + lm];
            v2f a = {ap[0], ap[1]};
            v2f b = {bq[0], bq[NN_D]};
            acc = __builtin_amdgcn_wmma_f32_16x16x4_f32(false, a, false, b,
                                                        (short)0, acc, false, false);
        }
        float* dp = &z2_ws[(m0 + moff) * NN_D + n0 + lm];
#pragma unroll
        for (int r = 0; r < 8; ++r) dp[r * NN_D] = acc[r];
    }
    __syncthreads();
    {
        int j = tid;
        float s = 0.f, sq = 0.f;
        for (int i = 0; i < NN_B; ++i) { float v = z2_ws[i * NN_D + j]; s += v; sq += v * v; }
        float mean = s * (1.0f / NN_B);
        float var  = sq * (1.0f / NN_B) - mean * mean;
        float sc   = g2[j] / sqrtf(var + BN_EPS);
        float be   = be2[j];
        for (int i = 0; i < NN_B; ++i) {
            float v = (z2_ws[i * NN_D + j] - mean) * sc + be;
            z2_ws[i * NN_D + j] = v > 0.f ? v : 0.f;
        }
    }
    __syncthreads();

    // ---- stage 5: out[64,10] = z2 @ W3[256,10] + b3 (one masked 16-col tile) ----
    if (wave < 4) {                          // wave-uniform branch: EXEC stays full
        int m0 = wave * 16;
        v8f acc = {0, 0, 0, 0, 0, 0, 0, 0};
        for (int k0 = 0; k0 < NN_D; k0 += 4) {
            const float* ap = &z2_ws[(m0 + lm) * NN_D + k0 + koff];
            v2f a = {ap[0], ap[1]};
            float bx = 0.f, by = 0.f;
            if (lm < NN_OUT) {
                const float* bq = &W3[(k0 + koff) * NN_OUT + lm];
                bx = bq[0]; by = bq[NN_OUT];
            }
            v2f b = {bx, by};
            acc = __builtin_amdgcn_wmma_f32_16x16x4_f32(false, a, false, b,
                                                        (short)0, acc, false, false);
        }
        if (lm < NN_OUT) {
            float bb = b3[lm];
#pragma unroll
            for (int r = 0; r < 8; ++r)
                out[(m0 + moff + r) * NN_OUT + lm] = acc[r] + bb;
        }
    }
}

// ---------------------------------------------------------------------------
extern "C" void kernel_launch(void* const* d_in, const int* in_sizes, int n_in,
                              void* d_out, int out_size, void* d_ws, size_t ws_size,
                              hipStream_t stream) {
    const int*   x     = (const int*)d_in[0];
    const int*   cv    = (const int*)d_in[1];
    const int*   batch = (const int*)d_in[2];
    const float* emb   = (const float*)d_in[3];
    const float* Wp    = (const float*)d_in[4];
    const float* bp    = (const float*)d_in[5];
    const float* W1    = (const float*)d_in[6];
    // d_in[7] = b1 (cancelled by BN mean subtraction)
    const float* g1    = (const float*)d_in[8];
    const float* be1   = (const float*)d_in[9];
    const float* W2    = (const float*)d_in[10];
    // d_in[11] = b2 (cancelled by BN mean subtraction)
    const float* g2    = (const float*)d_in[12];
    const float* be2   = (const float*)d_in[13];
    const float* W3    = (const float*)d_in[14];
    const float* b3    = (const float*)d_in[15];

    const int N = in_sizes[0];
    const int E = in_sizes[1];

    char* ws   = (char*)d_ws;
    int*  cnt  = (int*)ws;                                   // [N]
    int*  Cint = (int*)(ws + (size_t)N * 4);                 // [64*30], contiguous
    size_t off = (((size_t)N * 4 + (size_t)NN_B * NN_T * 4) + 127) & ~(size_t)127;
    float* g_ws  = (float*)(ws + off);                       // [64*128]
    float* z2_ws = (float*)(ws + off + (size_t)NN_B * NN_H * 4); // [64*256]

    int nz = N + NN_B * NN_T;   // cnt + Cint zeroed in one shot
    k_zero <<<(nz + 255) / 256, 256, 0, stream>>>(cnt, nz);
    k_count<<<(E  + 255) / 256, 256, 0, stream>>>(cv, cnt, E);
    k_hist <<<(N  + 255) / 256, 256, 0, stream>>>(x, batch, cnt, Cint, N);
    k_dense<<<1, 256, 0, stream>>>(Cint, emb, Wp, bp, W1, g1, be1,
                                   W2, g2, be2, W3, b3, g_ws, z2_ws, (float*)d_out);
}